// GCNN_26199300505723
// MI455X (gfx1250) — compile-verified
//
#include <hip/hip_runtime.h>

typedef __attribute__((ext_vector_type(2))) float v2f;
typedef __attribute__((ext_vector_type(8))) float v8f;
typedef __attribute__((ext_vector_type(4))) unsigned int u32x4;
typedef __attribute__((ext_vector_type(8))) int i32x8;
typedef __attribute__((ext_vector_type(4))) int i32x4;

#define NN   64000
#define FEAT 64
#define HID  512
#define OUTC 10
#define BSZ  64

// ---------------------------------------------------------------------------
// fp32 WMMA: D(16x16) = A(16x4) * B(4x16) + C   (exact fp32, CDNA5)
// A layout: lane l, elem j -> M = l%16, K = (l/16)*2 + j
// B layout: lane l, elem j -> N = l%16, K = (l/16)*2 + j
// C/D layout: lane l, vgpr v -> M = v + (l/16)*8, N = l%16
// ---------------------------------------------------------------------------
__device__ inline v8f wmma_f32_4(v2f a, v2f b, v8f c) {
  return __builtin_amdgcn_wmma_f32_16x16x4_f32(false, a, false, b,
                                               (short)0, c, false, false);
}

// -------------------------- elementwise helpers ----------------------------
__global__ void k_zero(float* __restrict__ p, int n) {
  int i = blockIdx.x * blockDim.x + threadIdx.x;
  if (i < n) p[i] = 0.f;
}

__global__ void k_negcopy(float* __restrict__ o, const float* __restrict__ a, int n) {
  int i = blockIdx.x * blockDim.x + threadIdx.x;
  if (i < n) o[i] = -a[i];
}

__global__ void k_degree(const int* __restrict__ src, float* __restrict__ deg, int E) {
  int e = blockIdx.x * blockDim.x + threadIdx.x;
  if (e < E) atomicAdd(&deg[src[e]], 1.0f);
}

__global__ void k_rsqrt(float* __restrict__ d, int n) {
  int i = blockIdx.x * blockDim.x + threadIdx.x;
  if (i < n) { float v = d[i]; d[i] = (v > 0.f) ? rsqrtf(v) : 0.f; }
}

__global__ void k_norm(const int* __restrict__ src, const int* __restrict__ dst,
                       const float* __restrict__ dis, float* __restrict__ nrm, int E) {
  int e = blockIdx.x * blockDim.x + threadIdx.x;
  if (e < E) nrm[e] = -dis[src[e]] * dis[dst[e]];
}

// out[dst[e]*F + f] += scale * nrm[e] * h[src[e]*F + f]   (L2-resident scatter)
template <int F>
__global__ void k_scatter(const float* __restrict__ h, const int* __restrict__ src,
                          const int* __restrict__ dst, const float* __restrict__ nrm,
                          float scale, float* __restrict__ out, int E) {
  int idx = blockIdx.x * blockDim.x + threadIdx.x;
  int e = idx / F;
  int f = idx % F;
  if (e >= E) return;
  float v = scale * nrm[e] * h[src[e] * F + f];
  atomicAdd(&out[dst[e] * F + f], v);
}

// Fin=1 Chebyshev combine (layer 1): rank-1 updates, no GEMM needed
__global__ void k_cheb1(const float* __restrict__ t0, const float* __restrict__ t1,
                        const float* __restrict__ t2, const float* __restrict__ W,
                        const float* __restrict__ bias, float* __restrict__ out) {
  int i = blockIdx.x * blockDim.x + threadIdx.x;
  if (i >= NN * FEAT) return;
  int n = i >> 6, f = i & 63;
  float v = t0[n] * W[f] + t1[n] * W[64 + f] + t2[n] * W[128 + f] + bias[f];
  out[i] = fmaxf(v, 0.f);
}

// ---------------------------------------------------------------------------
// Cheb GEMM (Fin=64): out[64000,64] = T0@W0 + T1@W1 + T2@W2 (+resid)(+bias)(relu)
// 4 waves / block, each wave owns one 16-row strip x all 64 cols (4 N-tiles).
// Weight block W (3*64*64 f32 = 48KB) is staged into LDS once per block by the
// Tensor Data Mover (TDM): wave 0 issues tensor_load_to_lds with a 1-D D#
// descriptor, waits on TENSORcnt, then all waves read B operands via LDS.
// The inline-asm escape after the barrier tells LLVM the TDM wrote Wlds, so
// the B-operand ds_loads cannot be folded to undef.
// ---------------------------------------------------------------------------
__global__ void __launch_bounds__(128) k_cheb_gemm64(
    const float* __restrict__ T0, const float* __restrict__ T1,
    const float* __restrict__ T2, const float* __restrict__ W,
    const float* __restrict__ bias, const float* __restrict__ resid,
    float* __restrict__ out, int do_relu) {
  __shared__ float Wlds[3 * FEAT * FEAT];            // 48 KB, LDS offset 0

  if (threadIdx.x < 32) {                            // wave 0 drives the TDM
    const unsigned long long ga = (unsigned long long)W;
    const unsigned elems = 3u * FEAT * FEAT;         // 12288 f32 elements
    u32x4 g0 = {};
    g0[0] = 1u;                                      // count=1 (valid user D#)
    g0[1] = 0u;                                      // lds_addr = 0 (Wlds base)
    g0[2] = (unsigned)(ga & 0xffffffffu);            // global_addr[31:0]
    g0[3] = (unsigned)((ga >> 32) & 0x01ffffffu)     // global_addr[56:32]
          | (2u << 30);                              // type = 2 ("image")
    i32x8 g1 = {};
    g1[0] = (int)(2u << 16);                         // data_size = 4B code 2
    g1[1] = (int)((elems & 0xffffu) << 16);          // tensor_dim0[15:0]
    g1[2] = (int)((elems >> 16) & 0xffffu)           // tensor_dim0[31:16]
          | (1 << 16);                               // tensor_dim1 = 1
    g1[3] = (int)((elems & 0xffffu) << 16);          // tile_dim0 = 12288
    g1[4] = 1;                                       // tile_dim1 = 1
    g1[5] = (int)elems;                              // tensor_dim0_stride
    g1[6] = 0;
    g1[7] = 0;
    i32x4 g2 = {};
    i32x4 g3 = {};
    i32x8 g4 = {};                                   // extra group (6-arg form)
    __builtin_amdgcn_tensor_load_to_lds(g0, g1, g2, g3, g4, 0);
    __builtin_amdgcn_s_wait_tensorcnt(0);            // s_wait_tensorcnt 0
  }
  __syncthreads();
  // The TDM wrote Wlds behind the compiler's back: escape the address with a
  // memory clobber so the LDS reads below are materialized as real ds_loads.
  asm volatile("" ::"v"(&Wlds[0]) : "memory");

  const int lane  = threadIdx.x & 31;
  const int wv    = threadIdx.x >> 5;
  const int mtile = blockIdx.x * 4 + wv;
  const int row0  = mtile * 16;
  const int ml    = lane & 15;   // M row (A) / N col (B,C)
  const int kg    = lane >> 4;   // K group

  v8f acc0 = {}, acc1 = {}, acc2 = {}, acc3 = {};

  const float* tb[3] = {T0, T1, T2};
#pragma unroll
  for (int t = 0; t < 3; ++t) {
    const float* __restrict__ Tb = tb[t];
    const float* Wt = &Wlds[t * FEAT * FEAT];
#pragma unroll 4
    for (int k0 = 0; k0 < FEAT; k0 += 4) {
      const int ka = k0 + kg * 2;
      v2f a = *(const v2f*)(Tb + (row0 + ml) * FEAT + ka);
      const float* w0 = Wt + ka * FEAT + ml;        // K=ka   row (LDS)
      const float* w1 = w0 + FEAT;                  // K=ka+1 row (LDS)
      v2f b0, b1, b2, b3;
      b0.x = w0[0];  b0.y = w1[0];
      b1.x = w0[16]; b1.y = w1[16];
      b2.x = w0[32]; b2.y = w1[32];
      b3.x = w0[48]; b3.y = w1[48];
      acc0 = wmma_f32_4(a, b0, acc0);
      acc1 = wmma_f32_4(a, b1, acc1);
      acc2 = wmma_f32_4(a, b2, acc2);
      acc3 = wmma_f32_4(a, b3, acc3);
    }
  }

  v8f accs[4] = {acc0, acc1, acc2, acc3};
#pragma unroll
  for (int nt = 0; nt < 4; ++nt) {
    const int n  = nt * 16 + ml;
    const float bv = bias[n];
#pragma unroll
    for (int v = 0; v < 8; ++v) {
      const int r = row0 + kg * 8 + v;
      float val = accs[nt][v] + bv;
      if (resid) val += resid[r * FEAT + n];
      if (do_relu) val = fmaxf(val, 0.f);
      out[r * FEAT + n] = val;
    }
  }
}

// ---------------------------------------------------------------------------
// lin1: G1[64,512] += H[64,64000] @ lin1_w[512,64000]^T   (K-split x125)
// grid = (ntile=32, mtile=4, ksplit=125), one wave per block.
// Partials accumulated with global_atomic_add_f32 into pre-zeroed G1.
// ---------------------------------------------------------------------------
__global__ void __launch_bounds__(32) k_lin1(const float* __restrict__ H,
                                             const float* __restrict__ Wm,
                                             float* __restrict__ G1) {
  const int lane = threadIdx.x;
  const int ml = lane & 15, kg = lane >> 4;
  const int ntile = blockIdx.x, mtile = blockIdx.y, ks = blockIdx.z;
  const float* __restrict__ Ar = H  + (mtile * 16 + ml) * (NN) + ks * 512 + kg * 2;
  const float* __restrict__ Br = Wm + (ntile * 16 + ml) * (NN) + ks * 512 + kg * 2;
  v8f acc = {};
  for (int k = 0; k < 512; k += 4) {
    if ((k & 63) == 0) {                 // stream 131MB weights: prefetch ahead
      __builtin_prefetch(Br + k + 128, 0, 0);
      __builtin_prefetch(Ar + k + 128, 0, 0);
    }
    v2f a = *(const v2f*)(Ar + k);
    v2f b = *(const v2f*)(Br + k);
    acc = wmma_f32_4(a, b, acc);
  }
  const int n = ntile * 16 + ml;
#pragma unroll
  for (int v = 0; v < 8; ++v)
    atomicAdd(&G1[(mtile * 16 + kg * 8 + v) * HID + n], acc[v]);
}

__global__ void k_bias_relu(float* __restrict__ g, const float* __restrict__ b, int n) {
  int i = blockIdx.x * blockDim.x + threadIdx.x;
  if (i < n) g[i] = fmaxf(g[i] + b[i & (HID - 1)], 0.f);
}

// lin2: G2[64,512] = relu(G1[64,512] @ lin2_w[512,512]^T + b). grid=(32,4), 1 wave.
__global__ void __launch_bounds__(32) k_lin2(const float* __restrict__ G1,
                                             const float* __restrict__ Wm,
                                             const float* __restrict__ bias,
                                             float* __restrict__ G2) {
  const int lane = threadIdx.x;
  const int ml = lane & 15, kg = lane >> 4;
  const int ntile = blockIdx.x, mtile = blockIdx.y;
  const float* __restrict__ Ar = G1 + (mtile * 16 + ml) * HID + kg * 2;
  const float* __restrict__ Br = Wm + (ntile * 16 + ml) * HID + kg * 2;
  v8f acc = {};
  for (int k = 0; k < HID; k += 4) {
    v2f a = *(const v2f*)(Ar + k);
    v2f b = *(const v2f*)(Br + k);
    acc = wmma_f32_4(a, b, acc);
  }
  const int n = ntile * 16 + ml;
  const float bv = bias[n];
#pragma unroll
  for (int v = 0; v < 8; ++v)
    G2[(mtile * 16 + kg * 8 + v) * HID + n] = fmaxf(acc[v] + bv, 0.f);
}

// lin3: tiny (64x10), plain dot products
__global__ void k_lin3(const float* __restrict__ G2, const float* __restrict__ Wm,
                       const float* __restrict__ bias, float* __restrict__ out) {
  int i = blockIdx.x * blockDim.x + threadIdx.x;
  if (i >= BSZ * OUTC) return;
  int g = i / OUTC, o = i % OUTC;
  const float* a = G2 + g * HID;
  const float* w = Wm + o * HID;
  float s = bias[o];
  for (int k = 0; k < HID; ++k) s += a[k] * w[k];
  out[i] = fmaxf(s, 0.f);
}

// ---------------------------------------------------------------------------
extern "C" void kernel_launch(void* const* d_in, const int* in_sizes, int n_in,
                              void* d_out, int out_size, void* d_ws, size_t ws_size,
                              hipStream_t stream) {
  (void)n_in; (void)out_size; (void)ws_size;
  const float* x    = (const float*)d_in[0];
  const int*   ei   = (const int*)d_in[1];
  const int    E    = in_sizes[1] / 2;           // 1,024,000 directed edges
  const int*   src  = ei;
  const int*   dst  = ei + E;
  const float* w1_1 = (const float*)d_in[3];  const float* b1_1 = (const float*)d_in[4];
  const float* w1_2 = (const float*)d_in[5];  const float* b1_2 = (const float*)d_in[6];
  const float* w1_3 = (const float*)d_in[7];  const float* b1_3 = (const float*)d_in[8];
  const float* w2_1 = (const float*)d_in[9];  const float* b2_1 = (const float*)d_in[10];
  const float* w2_2 = (const float*)d_in[11]; const float* b2_2 = (const float*)d_in[12];
  const float* w3_1 = (const float*)d_in[13]; const float* b3_1 = (const float*)d_in[14];
  const float* l1w  = (const float*)d_in[15]; const float* l1b  = (const float*)d_in[16];
  const float* l2w  = (const float*)d_in[17]; const float* l2b  = (const float*)d_in[18];
  const float* l3w  = (const float*)d_in[19]; const float* l3b  = (const float*)d_in[20];

  float* wsp  = (float*)d_ws;
  float* deg  = wsp; wsp += NN;          // becomes dis in place
  float* nrm  = wsp; wsp += E;
  float* H    = wsp; wsp += NN * FEAT;
  float* TMP  = wsp; wsp += NN * FEAT;
  float* T1   = wsp; wsp += NN * FEAT;
  float* T2   = wsp; wsp += NN * FEAT;
  float* OB   = wsp; wsp += NN * FEAT;
  float* G1   = wsp; wsp += BSZ * HID;
  float* G2   = wsp; wsp += BSZ * HID;

  auto cdiv = [](int a, int b) { return (a + b - 1) / b; };

  // ---- edge norm: dis = deg^{-1/2}; norm = -dis[src]*dis[dst] (L2-resident)
  k_zero  <<<cdiv(NN, 256), 256, 0, stream>>>(deg, NN);
  k_degree<<<cdiv(E, 256), 256, 0, stream>>>(src, deg, E);
  k_rsqrt <<<cdiv(NN, 256), 256, 0, stream>>>(deg, NN);
  k_norm  <<<cdiv(E, 256), 256, 0, stream>>>(src, dst, deg, nrm, E);

  // ---- layer 1 (Fin=1): Tx1 = L x; Tx2 = 2 L Tx1 - x; combine + relu
  k_zero<<<cdiv(NN, 256), 256, 0, stream>>>(T1, NN);
  k_scatter<1><<<cdiv(E, 256), 256, 0, stream>>>(x, src, dst, nrm, 1.f, T1, E);
  k_negcopy<<<cdiv(NN, 256), 256, 0, stream>>>(T2, x, NN);
  k_scatter<1><<<cdiv(E, 256), 256, 0, stream>>>(T1, src, dst, nrm, 2.f, T2, E);
  k_cheb1<<<cdiv(NN * FEAT, 256), 256, 0, stream>>>(x, T1, T2, w1_1, b1_1, H);
  (void)hipMemcpyAsync(TMP, H, sizeof(float) * NN * FEAT, hipMemcpyDeviceToDevice, stream);

  float* Hp = H;
  float* Op = OB;
  auto cheb64 = [&](const float* in, const float* W, const float* bias,
                    const float* resid, int relu, float* outb) {
    k_zero<<<cdiv(NN * FEAT, 256), 256, 0, stream>>>(T1, NN * FEAT);
    k_scatter<FEAT><<<cdiv(E * FEAT, 256), 256, 0, stream>>>(in, src, dst, nrm, 1.f, T1, E);
    k_negcopy<<<cdiv(NN * FEAT, 256), 256, 0, stream>>>(T2, in, NN * FEAT);
    k_scatter<FEAT><<<cdiv(E * FEAT, 256), 256, 0, stream>>>(T1, src, dst, nrm, 2.f, T2, E);
    k_cheb_gemm64<<<NN / 64, 128, 0, stream>>>(in, T1, T2, W, bias, resid, outb, relu);
  };
  auto swp = [&]() { float* t = Hp; Hp = Op; Op = t; };

  cheb64(Hp, w1_2, b1_2, nullptr, 1, Op); swp();           // relu(cheb)
  cheb64(Hp, w1_3, b1_3, TMP,     0, Op); swp();           // cheb + tmp
  (void)hipMemcpyAsync(TMP, Hp, sizeof(float) * NN * FEAT, hipMemcpyDeviceToDevice, stream);
  cheb64(Hp, w2_1, b2_1, nullptr, 1, Op); swp();           // relu(cheb)
  cheb64(Hp, w2_2, b2_2, TMP,     0, Op); swp();           // cheb + tmp
  cheb64(Hp, w3_1, b3_1, nullptr, 1, Op); swp();           // relu(cheb)

  // ---- readout: (64,64000) view of Hp is just its flat layout
  k_zero<<<cdiv(BSZ * HID, 256), 256, 0, stream>>>(G1, BSZ * HID);
  k_lin1<<<dim3(32, 4, 125), 32, 0, stream>>>(Hp, l1w, G1);
  k_bias_relu<<<cdiv(BSZ * HID, 256), 256, 0, stream>>>(G1, l1b, BSZ * HID);
  k_lin2<<<dim3(32, 4), 32, 0, stream>>>(G1, l2w, l2b, G2);
  k_lin3<<<cdiv(BSZ * OUTC, 64), 64, 0, stream>>>(G2, l3w, l3b, (float*)d_out);
}